// InverseMoE_30691836297576
// MI455X (gfx1250) — compile-verified
//
#include <hip/hip_runtime.h>

#define DIMN 1024
#define NEXP 20
#define NFRQ 3000
#define TOPK 5
#define BATCH 64

typedef __attribute__((ext_vector_type(2))) float v2f;
typedef __attribute__((ext_vector_type(4))) float v4f;
typedef __attribute__((ext_vector_type(8))) float v8f;

// ---------------------------------------------------------------------------
// Kernel 1: router GEMM (fp32 WMMA 16x16x4) + bias + top-5 expert selection.
// One block, 8 waves. Wave w handles row tile (w>>1)*16 and expert tile
// (w&1)*16 (experts 20..31 are -inf padding). Softmax is skipped: top-k of
// logits == top-k of softmax(logits).
// ---------------------------------------------------------------------------
__global__ __launch_bounds__(256) void router_topk_kernel(
    const float* __restrict__ cls, const float* __restrict__ W,
    const float* __restrict__ bias, int* __restrict__ wsExp)
{
    __shared__ float slog[BATCH * 32];

    const int tid  = threadIdx.x;
    const int wave = tid >> 5;
    const int lane = tid & 31;
    const int half = lane >> 4;   // 0: lanes 0-15, 1: lanes 16-31
    const int l15  = lane & 15;
    const int m0   = (wave >> 1) * 16;   // row tile base (0,16,32,48)
    const int n0   = (wave & 1) * 16;    // expert tile base (0,16)

    const int n   = n0 + l15;                 // this lane's expert column
    const int nc  = (n < NEXP) ? n : (NEXP - 1); // clamped (memory-safe)
    const int m   = m0 + l15;                 // this lane's A row

    const float* __restrict__ arow = cls + m * DIMN;

    v8f c = {};
    #pragma unroll 4
    for (int k0 = 0; k0 < DIMN; k0 += 4) {
        const int kk = k0 + 2 * half;        // per-lane K pair: kk, kk+1
        // A 16x4 f32: VGPR0 = K{0,2}, VGPR1 = K{1,3} split by lane half
        v2f a = *(const v2f*)(arow + kk);    // 8B aligned (kk even)
        // B 4x16 f32: lanes = N, same K-pair split by lane half
        v2f b;
        b.x = W[kk * NEXP + nc];
        b.y = W[(kk + 1) * NEXP + nc];
        c = __builtin_amdgcn_wmma_f32_16x16x4_f32(
                /*neg_a=*/false, a, /*neg_b=*/false, b,
                /*c_mod=*/(short)0, c, /*reuse_a=*/false, /*reuse_b=*/false);
    }

    // C/D layout: VGPR j, lane -> M = j + 8*half, N = lane&15
    const bool  valid = (n < NEXP);
    const float bv    = valid ? bias[n] : 0.0f;
    #pragma unroll
    for (int j = 0; j < 8; ++j) {
        const int row = m0 + j + 8 * half;
        slog[row * 32 + n] = valid ? (c[j] + bv) : -1e30f;
    }
    __syncthreads();

    // Top-5 per row: strict '>' scan reproduces jax top_k lowest-index
    // tie-breaking. One thread per row.
    if (tid < BATCH) {
        float* rl = slog + tid * 32;
        #pragma unroll
        for (int t = 0; t < TOPK; ++t) {
            float best = -3e38f; int bi = 0;
            for (int e = 0; e < NEXP; ++e) {
                const float v = rl[e];
                if (v > best) { best = v; bi = e; }
            }
            wsExp[tid * TOPK + t] = bi;
            rl[bi] = -3e38f;   // exclude from next pick
        }
    }
}

// ---------------------------------------------------------------------------
// Kernel 2: zero-fill the 256MB output with non-temporal 128-bit stores.
// This is the roofline-dominant step (~11us at 23.3 TB/s).
// ---------------------------------------------------------------------------
__global__ __launch_bounds__(256) void fill_zero_kernel(
    float* __restrict__ out, long long n)
{
    const long long stride = (long long)gridDim.x * blockDim.x;
    const long long base   = (long long)blockIdx.x * blockDim.x + threadIdx.x;
    const long long n4     = n >> 2;
    v4f z = {};
    for (long long i = base; i < n4; i += stride)
        __builtin_nontemporal_store(z, (v4f*)out + i);
    if (base == 0)
        for (long long i = n4 << 2; i < n; ++i) out[i] = 0.0f;
}

// ---------------------------------------------------------------------------
// Kernel 3: scatter ones. One block per (row, topk-slot): 3000 indices each.
// ---------------------------------------------------------------------------
__global__ __launch_bounds__(256) void scatter_ones_kernel(
    float* __restrict__ out, const int* __restrict__ wsExp,
    const int* __restrict__ listIdx)
{
    const int slot = blockIdx.x;          // 0 .. BATCH*TOPK-1
    const int row  = slot / TOPK;
    const int e    = wsExp[slot];
    const int* __restrict__ src = listIdx + e * NFRQ;
    float* __restrict__ dst = out + (long long)row * (DIMN * DIMN);
    for (int i = threadIdx.x; i < NFRQ; i += blockDim.x)
        dst[src[i]] = 1.0f;
}

extern "C" void kernel_launch(void* const* d_in, const int* in_sizes, int n_in,
                              void* d_out, int out_size, void* d_ws, size_t ws_size,
                              hipStream_t stream) {
    const float* cls  = (const float*)d_in[0];   // [64,1024] f32
    const float* W    = (const float*)d_in[1];   // [1024,20] f32
    const float* bias = (const float*)d_in[2];   // [20] f32
    const int*   li   = (const int*)d_in[3];     // [20,3000] i32
    float* out  = (float*)d_out;                 // [64,1024,1024] f32
    int* wsExp  = (int*)d_ws;                    // 320 ints of scratch

    router_topk_kernel<<<1, 256, 0, stream>>>(cls, W, bias, wsExp);
    fill_zero_kernel<<<8192, 256, 0, stream>>>(out, (long long)out_size);
    scatter_ones_kernel<<<BATCH * TOPK, 256, 0, stream>>>(out, wsExp, li);
}